// myHIN_75900662054957
// MI455X (gfx1250) — compile-verified
//
#include <hip/hip_runtime.h>
#include <math.h>

// ---------------------------------------------------------------------------
// HIN recommender forward for MI455X (gfx1250, wave32, WMMA).
// Only user_final . course_emb reaches the output; the course-neighbor branch
// in the reference is dead code and is skipped.
// All GEMMs use V_WMMA_F32_16X16X4_F32 (full f32 precision, D=16 tiles).
// ---------------------------------------------------------------------------

typedef float v2f __attribute__((ext_vector_type(2)));
typedef float v8f __attribute__((ext_vector_type(8)));

#define DD   16
#define LSQ  200   // sequence length
#define LTE  10    // teachers per user
#define LSC  5     // schools per user
#define NCC  100   // courses per row
#define LP   208   // padded rows (13 * 16)
#define NTI  13    // 16-row tiles
#define XS   20    // padded LDS row stride for x / Q (bank-conflict avoidance)
#define TS   17    // per-wave 16x16 tile stride
#define NWAVE 8

// Per-wave LDS ordering fence: DS ops of one wave complete in order, so after
// s_wait_dscnt==0 every lane's prior LDS write is visible to every lane's
// subsequent LDS read. wave_barrier stops compiler reordering.
__device__ __forceinline__ void lds_fence() {
  asm volatile("s_wait_dscnt 0" ::: "memory");
  __builtin_amdgcn_wave_barrier();
}

// D(16x16,f32) += A(16x4,f32) * B(4x16,f32)
__device__ __forceinline__ v8f wmma4(float a0, float a1, float b0, float b1, v8f c) {
  v2f a; a.x = a0; a.y = a1;
  v2f b; b.x = b0; b.y = b1;
  return __builtin_amdgcn_wmma_f32_16x16x4_f32(false, a, false, b, (short)0, c,
                                               false, false);
}

__global__ __launch_bounds__(256) void hin_fwd(
    const int*   __restrict__ user_idx,       // (B,1)
    const int*   __restrict__ user_sequence,  // (B,200)
    const int*   __restrict__ user_teachers,  // (B,10)
    const int*   __restrict__ user_school,    // (B,5)
    const float* __restrict__ user_len_teacher, // (B,1)
    const float* __restrict__ user_len_school,  // (B,1)
    const int*   __restrict__ course_set,     // (B,100)
    const float* __restrict__ user_table,     // (100001,16)
    const float* __restrict__ course_table,   // (10001,16)
    const float* __restrict__ teacher_table,  // (5001,16)
    const float* __restrict__ school_table,   // (3001,16)
    const float* __restrict__ Qw, const float* __restrict__ Qb,
    const float* __restrict__ Kw, const float* __restrict__ Kb,
    const float* __restrict__ g1w, const float* __restrict__ g1b,
    const float* __restrict__ g2w, const float* __restrict__ g2b,
    float* __restrict__ out)                  // (B,100)
{
  __shared__ float sh_x [LP * XS];      // gathered sequence embeddings (padded)
  __shared__ float sh_q [LP * XS];      // relu(x Qw + Qb)
  __shared__ float sh_kr[LP * DD];      // relu(x Kw + Kb), flat stride 16 (reshape view)
  __shared__ float sh_w [544];          // Qw | Kw | Qb | Kb
  __shared__ float sh_t [NWAVE * 16 * TS]; // per-wave 16x16 staging tile
  __shared__ float sh_c [NWAVE * 16];   // per-wave per-row broadcast scalars
  __shared__ float sh_seq[DD];
  __shared__ float sh_h  [2 * DD];
  __shared__ float sh_hid[DD];
  __shared__ float sh_uf [DD];

  const int b    = blockIdx.x;
  const int tid  = threadIdx.x;
  // Force wave index into an SGPR so all wave-uniform control flow / addressing
  // derived from it is scalar (no EXEC-mask divergence on stores).
  const int wave = __builtin_amdgcn_readfirstlane(tid >> 5);
  const int lane = tid & 31;
  const int ln16 = lane & 15;
  const int half = lane >> 4;   // 0: lanes 0-15, 1: lanes 16-31
  const int off  = half * 8;    // D-frag: VGPR r holds row r+off
  const int q0   = half * 2;    // A/B-frag: this half covers K sub-cols {q0,q0+1}

  // ---- Phase 0: init + weight preload + gather sequence embeddings --------
  if (tid < DD) sh_seq[tid] = 0.0f;
  sh_w[tid]       = Qw[tid & 255];   // tid<256 always
  sh_w[256 + tid] = Kw[tid & 255];
  if (tid < DD) { sh_w[512 + tid] = Qb[tid]; sh_w[528 + tid] = Kb[tid]; }
  // float4 gather: 4 threads per sequence row, b128 loads / b128 LDS stores
  for (int e = tid; e < LP * 4; e += 256) {
    const int l = e >> 2, dq = e & 3;
    float4 v = make_float4(0.f, 0.f, 0.f, 0.f);
    if (l < LSQ) {
      const int ci = user_sequence[b * LSQ + l];
      v = ((const float4*)course_table)[ci * 4 + dq];
    }
    *(float4*)&sh_x[l * XS + dq * 4] = v;   // (l*20 + dq*4)*4B is 16B aligned
  }
  __syncthreads();

  // ---- Phase 1: Q = relu(x Qw + Qb); Kr = relu(x Kw + Kb) via WMMA --------
  for (int job = wave; job < 2 * NTI; job += NWAVE) {   // job is SGPR-uniform
    const int mat = (job >= NTI) ? 1 : 0;   // 0 -> Q, 1 -> K
    const int ti  = job - mat * NTI;
    const float* W   = sh_w + mat * 256;
    const float* Bv  = sh_w + 512 + mat * 16;
    float* dst       = mat ? sh_kr : sh_q;  // single-store pointer select
    const int stride = mat ? DD : XS;
    v8f f = {0.f,0.f,0.f,0.f,0.f,0.f,0.f,0.f};
#pragma unroll
    for (int kc = 0; kc < 4; ++kc) {
      const int k0 = kc * 4 + q0;
      const float a0 = sh_x[(ti * 16 + ln16) * XS + k0];
      const float a1 = sh_x[(ti * 16 + ln16) * XS + k0 + 1];
      const float b0 = W[k0 * DD + ln16];
      const float b1 = W[(k0 + 1) * DD + ln16];
      f = wmma4(a0, a1, b0, b1, f);
    }
    const float bias = Bv[ln16];
#pragma unroll
    for (int r = 0; r < 8; ++r) {
      const int row = ti * 16 + r + off;
      float v = f[r] + bias;
      v = v > 0.f ? v : 0.f;
      dst[row * stride + ln16] = v;
    }
  }
  __syncthreads();

  // ---- Phase 2: attention rows, flash-style online softmax ----------------
  float* T = sh_t + wave * 16 * TS;
  float* C = sh_c + wave * 16;

  for (int ti = wave; ti < NTI; ti += NWAVE) {
    v8f acc = {0.f,0.f,0.f,0.f,0.f,0.f,0.f,0.f};
    float m_run = -3.0e38f;   // running row max (owned by lanes 0-15, row=lane)
    float s_run = 0.0f;       // running row sum of exp

    // Q A-fragment is invariant across J: hoist out of the loop (the fences'
    // memory clobbers would otherwise force a reload every iteration).
    float qa0[4], qa1[4];
#pragma unroll
    for (int kc = 0; kc < 4; ++kc) {
      const int k0 = kc * 4 + q0;
      qa0[kc] = sh_q[(ti * 16 + ln16) * XS + k0];
      qa1[kc] = sh_q[(ti * 16 + ln16) * XS + k0 + 1];
    }

    for (int J = 0; J < NTI; ++J) {
      // s tile = Q[ti] (16x16) * Kview[:, J*16..] (16x16), K split into 4x K=4
      v8f f = {0.f,0.f,0.f,0.f,0.f,0.f,0.f,0.f};
#pragma unroll
      for (int kc = 0; kc < 4; ++kc) {
        const int k0 = kc * 4 + q0;
        // Kview[k][j] = krflat[k*200 + j]  (the reference's reshape, not a transpose)
        const float b0 = sh_kr[k0 * LSQ + J * 16 + ln16];
        const float b1 = sh_kr[(k0 + 1) * LSQ + J * 16 + ln16];
        f = wmma4(qa0[kc], qa1[kc], b0, b1, f);
      }
      const int gcol = J * 16 + ln16;
#pragma unroll
      for (int r = 0; r < 8; ++r) {
        const int rr = r + off;
        float v = f[r] * 0.25f;                    // / sqrt(16)
        if (J == ti && rr == ln16) v = 1e-32f;     // diagonal mask
        if (gcol >= LSQ)           v = -3.0e38f;   // padded columns -> exp()=0
        T[rr * TS + ln16] = v;
      }
      lds_fence();

      // per-row softmax statistics update (lanes 0-15, one row each)
      if (lane < 16) {
        const int grow = ti * 16 + lane;
        if (grow < LSQ) {
          float vals[16];
          float mt = -3.0e38f;
#pragma unroll
          for (int n = 0; n < 16; ++n) { vals[n] = T[lane * TS + n]; mt = fmaxf(mt, vals[n]); }
          const float nm   = fmaxf(m_run, mt);
          const float corr = __expf(m_run - nm);
          float rs = 0.f;
#pragma unroll
          for (int n = 0; n < 16; ++n) {
            const float p = __expf(vals[n] - nm);
            T[lane * TS + n] = p;
            rs += p;
          }
          s_run = s_run * corr + rs;
          m_run = nm;
          C[lane] = corr;
        } else {
          // padded row: contributes nothing, keep accumulator untouched
#pragma unroll
          for (int n = 0; n < 16; ++n) T[lane * TS + n] = 0.f;
          C[lane] = 1.0f;
        }
      }
      lds_fence();

      // rescale running accumulator, then acc += P(16x16) * x[J] (16x16)
#pragma unroll
      for (int r = 0; r < 8; ++r) acc[r] = acc[r] * C[r + off];
#pragma unroll
      for (int kc = 0; kc < 4; ++kc) {
        const int k0 = kc * 4 + q0;
        const float a0 = T[ln16 * TS + k0];
        const float a1 = T[ln16 * TS + k0 + 1];
        const float b0 = sh_x[(J * 16 + k0) * XS + ln16];
        const float b1 = sh_x[(J * 16 + k0 + 1) * XS + ln16];
        acc = wmma4(a0, a1, b0, b1, acc);
      }
      lds_fence();   // T is reused next J
    }

    // normalize rows (1/s_run; 0 for padded rows) and fold into column sums
    if (lane < 16) {
      const int grow = ti * 16 + lane;
      C[lane] = (grow < LSQ) ? (1.0f / s_run) : 0.f;
    }
    lds_fence();
    float partial = 0.f;
#pragma unroll
    for (int r = 0; r < 8; ++r) partial += acc[r] * C[r + off];
    atomicAdd(&sh_seq[ln16], partial);   // ds_add_f32, seq_rep column sums
    lds_fence();
  }
  __syncthreads();

  // ---- Phase 3: user-side aggregation + 2-layer MLP -----------------------
  if (tid < DD) {
    const int d = tid;
    float ut = 0.f;
#pragma unroll
    for (int t = 0; t < LTE; ++t) ut += teacher_table[user_teachers[b * LTE + t] * DD + d];
    ut /= user_len_teacher[b];
    float us = 0.f;
#pragma unroll
    for (int t = 0; t < LSC; ++t) us += school_table[user_school[b * LSC + t] * DD + d];
    us /= user_len_school[b];
    const float seq = sh_seq[d] / (float)LSQ;     // a.mean(axis=1)
    const float agg = (seq + ut + us) / 3.0f;
    sh_h[d]      = user_table[user_idx[b] * DD + d];
    sh_h[DD + d] = agg;
  }
  __syncthreads();
  if (tid < DD) {
    float a = g1b[tid];
#pragma unroll
    for (int k = 0; k < 2 * DD; ++k) a += sh_h[k] * g1w[k * DD + tid];
    sh_hid[tid] = a > 0.f ? a : 0.f;
  }
  __syncthreads();
  if (tid < DD) {
    float a = g2b[tid];
#pragma unroll
    for (int j = 0; j < DD; ++j) a += sh_hid[j] * g2w[j * DD + tid];
    sh_uf[tid] = a;
  }
  __syncthreads();

  // ---- Phase 4: out[b,c] = user_final . course_emb[b,c] -------------------
  if (tid < NCC) {
    const int ci = course_set[b * NCC + tid];
    const float4* ce = (const float4*)course_table + ci * 4;
    float o = 0.f;
#pragma unroll
    for (int dq = 0; dq < 4; ++dq) {
      const float4 v = ce[dq];
      o += sh_uf[dq * 4 + 0] * v.x + sh_uf[dq * 4 + 1] * v.y +
           sh_uf[dq * 4 + 2] * v.z + sh_uf[dq * 4 + 3] * v.w;
    }
    out[b * NCC + tid] = o;
  }
}

extern "C" void kernel_launch(void* const* d_in, const int* in_sizes, int n_in,
                              void* d_out, int out_size, void* d_ws, size_t ws_size,
                              hipStream_t stream) {
  (void)n_in; (void)out_size; (void)d_ws; (void)ws_size;
  const int*   user_idx         = (const int*)  d_in[0];
  const int*   user_sequence    = (const int*)  d_in[1];
  const int*   user_teachers    = (const int*)  d_in[2];
  const int*   user_school      = (const int*)  d_in[3];
  // d_in[4] user_len_seq: unused by the reference output
  const float* user_len_teacher = (const float*)d_in[5];
  const float* user_len_school  = (const float*)d_in[6];
  const int*   course_set       = (const int*)  d_in[7];
  // d_in[8..12]: course-neighbor branch is dead code w.r.t. the output
  const float* user_table       = (const float*)d_in[13];
  const float* course_table     = (const float*)d_in[14];
  const float* teacher_table    = (const float*)d_in[15];
  const float* school_table     = (const float*)d_in[16];
  const float* Qw  = (const float*)d_in[17];
  const float* Qb  = (const float*)d_in[18];
  const float* Kw  = (const float*)d_in[19];
  const float* Kb  = (const float*)d_in[20];
  const float* g1w = (const float*)d_in[21];
  const float* g1b = (const float*)d_in[22];
  const float* g2w = (const float*)d_in[23];
  const float* g2b = (const float*)d_in[24];

  const int B = in_sizes[0];   // 1024 batch rows, one workgroup each
  hin_fwd<<<B, 256, 0, stream>>>(user_idx, user_sequence, user_teachers,
                                 user_school, user_len_teacher, user_len_school,
                                 course_set, user_table, course_table,
                                 teacher_table, school_table,
                                 Qw, Qb, Kw, Kb, g1w, g1b, g2w, g2b,
                                 (float*)d_out);
}